// LSTMADecoder_74955769250084
// MI455X (gfx1250) — compile-verified
//
#include <hip/hip_runtime.h>
#include <math.h>

// ---------------------------------------------------------------------------
// LSTM attention decoder step, MI455X (gfx1250, wave32).
// HBM-bandwidth bound: ~340 MB fp32 weights -> ~14.6 us floor at 23.3 TB/s.
// GEMV core: V_WMMA_F32_16X16X4_F32, 16 rows/wave, b128 weight loads feeding
// two WMMAs per load, split-K partials for enough waves to fill the HBM pipe.
// Grid encodes (row-block, k-segment) as (blockIdx.x, blockIdx.y): no div/mod.
// ---------------------------------------------------------------------------

typedef __attribute__((ext_vector_type(2))) float v2f;
typedef __attribute__((ext_vector_type(4))) float v4f;
typedef __attribute__((ext_vector_type(8))) float v8f;

#define TPB 256
#define Hdim 1024
#define Ldim 2048
#define Vdim 50257
#define ACH  64        // split-K chunks for attn_applied

// ---------------------------------------------------------------------------
// prep: gather embedding row, build cat1 = [emb, h0], seed cat2[0:H]=emb,
// copy h0/c0 into working h/c buffers.
// ---------------------------------------------------------------------------
__global__ void __launch_bounds__(TPB)
prep_kernel(const int* __restrict__ tok, const float* __restrict__ emb_table,
            const float* __restrict__ h0, const float* __restrict__ c0,
            float* __restrict__ cat1, float* __restrict__ cat2,
            float* __restrict__ hbuf, float* __restrict__ cbuf) {
  int i = blockIdx.x * TPB + threadIdx.x;
  if (i >= Hdim) return;
  int t = tok[0];
  float e = emb_table[(size_t)t * Hdim + i];
  cat1[i] = e;
  cat1[Hdim + i] = h0[i];
  cat2[i] = e;
  hbuf[i] = h0[i];
  cbuf[i] = c0[i];
}

// ---------------------------------------------------------------------------
// Split-K WMMA GEMV partial:
//   P[seg*M + m] = dot(W1[m, k0:k0+Kseg], x1[k0:]) (+ same for W2/x2)
// grid = dim3(nRB, S): blockIdx.x = row-block (128 rows), blockIdx.y = seg.
// Per wave: 16 rows. Each lane does one b128 weight load per 8-k step:
//   lanes 0-15 hold rows at k..k+3, lanes 16-31 same rows at k+4..k+7.
// WMMA #1 logical K = physical {k,k+1,k+4,k+5}; WMMA #2 = {k+2,k+3,k+6,k+7};
// B carries x (same remap) in column N=0 only (lanes 0 and 16 non-zero).
// D column 0: lane 0 = rows 0..7 (acc[0..7]), lane 16 = rows 8..15.
// Requires Kseg multiple of 8 and <= 512 (all call sites comply).
// ---------------------------------------------------------------------------
__global__ void __launch_bounds__(TPB)
gemv_wmma_partial(const float* __restrict__ W1, const float* __restrict__ x1,
                  const float* __restrict__ W2, const float* __restrict__ x2,
                  float* __restrict__ P, int M, int K, int Kseg) {
  __shared__ float sx1[512];
  __shared__ float sx2[512];
  const int rb  = blockIdx.x;
  const int seg = blockIdx.y;
  const int k0  = seg * Kseg;

  // stage the x segment(s) into LDS with b128 loads
  {
    const int nv = Kseg >> 2;               // <= 128 < TPB: single pass
    if (threadIdx.x < nv)
      ((v4f*)sx1)[threadIdx.x] = ((const v4f*)(x1 + k0))[threadIdx.x];
    if (W2) {
      if (threadIdx.x < nv)
        ((v4f*)sx2)[threadIdx.x] = ((const v4f*)(x2 + k0))[threadIdx.x];
    }
  }
  __syncthreads();

  const int wave = threadIdx.x >> 5;        // wave32
  const int lane = threadIdx.x & 31;
  const int row0 = (rb * 8 + wave) * 16;
  if (row0 >= M) return;                    // wave-uniform: EXEC stays all-ones

  int r = row0 + (lane & 15); if (r > M - 1) r = M - 1;   // clamp tail rows
  const int koff = (lane >> 4) * 4;                        // 0 or 4
  const bool keep = (lane & 15) == 0;                      // lanes 0,16 carry x

  v8f acc = {};
  {
    const float* Wp = W1 + (size_t)r * K + k0 + koff;
#pragma unroll 4
    for (int k = 0; k < Kseg; k += 8) {
      v4f a  = __builtin_nontemporal_load((const v4f*)(Wp + k)); // stream once
      v4f xv = *(const v4f*)(sx1 + k + koff);
      v2f a1; a1.x = a.x; a1.y = a.y;
      v2f a2; a2.x = a.z; a2.y = a.w;
      v2f bA; bA.x = keep ? xv.x : 0.0f; bA.y = keep ? xv.y : 0.0f;
      v2f bB; bB.x = keep ? xv.z : 0.0f; bB.y = keep ? xv.w : 0.0f;
      acc = __builtin_amdgcn_wmma_f32_16x16x4_f32(false, a1, false, bA,
                                                  (short)0, acc, false, false);
      acc = __builtin_amdgcn_wmma_f32_16x16x4_f32(false, a2, false, bB,
                                                  (short)0, acc, false, false);
    }
  }
  if (W2) {
    const float* Wp = W2 + (size_t)r * K + k0 + koff;
#pragma unroll 4
    for (int k = 0; k < Kseg; k += 8) {
      v4f a  = __builtin_nontemporal_load((const v4f*)(Wp + k));
      v4f xv = *(const v4f*)(sx2 + k + koff);
      v2f a1; a1.x = a.x; a1.y = a.y;
      v2f a2; a2.x = a.z; a2.y = a.w;
      v2f bA; bA.x = keep ? xv.x : 0.0f; bA.y = keep ? xv.y : 0.0f;
      v2f bB; bB.x = keep ? xv.z : 0.0f; bB.y = keep ? xv.w : 0.0f;
      acc = __builtin_amdgcn_wmma_f32_16x16x4_f32(false, a1, false, bA,
                                                  (short)0, acc, false, false);
      acc = __builtin_amdgcn_wmma_f32_16x16x4_f32(false, a2, false, bB,
                                                  (short)0, acc, false, false);
    }
  }

  const int base = (lane == 0) ? row0 : ((lane == 16) ? row0 + 8 : -1);
  if (base >= 0) {
    float* Pp = P + (size_t)seg * M;
#pragma unroll
    for (int j = 0; j < 8; ++j) {
      int rr = base + j;
      if (rr < M) Pp[rr] = acc[j];
    }
  }
}

// y[m] = act( sum_s P[s*M+m] + b1[m] + b2[m] )   fixed order => deterministic
__global__ void __launch_bounds__(TPB)
gemv_finalize_kernel(const float* __restrict__ P, const float* __restrict__ b1,
                     const float* __restrict__ b2, float* __restrict__ y,
                     int M, int S, int relu) {
  int m = blockIdx.x * TPB + threadIdx.x;
  if (m >= M) return;
  float v = 0.0f;
  for (int s = 0; s < S; ++s) v += P[(size_t)s * M + m];
  if (b1) v += b1[m];
  if (b2) v += b2[m];
  if (relu) v = fmaxf(v, 0.0f);
  y[m] = v;
}

// ---------------------------------------------------------------------------
// deterministic block reductions for log_softmax
// ---------------------------------------------------------------------------
__global__ void __launch_bounds__(TPB)
reduce_max_kernel(const float* __restrict__ x, int n, float* __restrict__ blk) {
  __shared__ float s[TPB];
  float m = -INFINITY;
  for (int i = blockIdx.x * TPB + threadIdx.x; i < n; i += gridDim.x * TPB)
    m = fmaxf(m, x[i]);
  s[threadIdx.x] = m; __syncthreads();
  for (int off = TPB / 2; off > 0; off >>= 1) {
    if (threadIdx.x < off) s[threadIdx.x] = fmaxf(s[threadIdx.x], s[threadIdx.x + off]);
    __syncthreads();
  }
  if (threadIdx.x == 0) blk[blockIdx.x] = s[0];
}

__global__ void __launch_bounds__(TPB)
reduce_sumexp_kernel(const float* __restrict__ x, int n,
                     const float* __restrict__ maxp, float* __restrict__ blk) {
  __shared__ float s[TPB];
  float mx = maxp[0];
  float a = 0.0f;
  for (int i = blockIdx.x * TPB + threadIdx.x; i < n; i += gridDim.x * TPB)
    a += expf(x[i] - mx);
  s[threadIdx.x] = a; __syncthreads();
  for (int off = TPB / 2; off > 0; off >>= 1) {
    if (threadIdx.x < off) s[threadIdx.x] += s[threadIdx.x + off];
    __syncthreads();
  }
  if (threadIdx.x == 0) blk[blockIdx.x] = s[0];
}

// op: 0 = max, 1 = sum   (single block)
__global__ void __launch_bounds__(TPB)
reduce_final_kernel(const float* __restrict__ in, int n, float* __restrict__ out, int op) {
  __shared__ float s[TPB];
  float v = (op == 0) ? -INFINITY : 0.0f;
  for (int i = threadIdx.x; i < n; i += TPB)
    v = (op == 0) ? fmaxf(v, in[i]) : (v + in[i]);
  s[threadIdx.x] = v; __syncthreads();
  for (int off = TPB / 2; off > 0; off >>= 1) {
    if (threadIdx.x < off)
      s[threadIdx.x] = (op == 0) ? fmaxf(s[threadIdx.x], s[threadIdx.x + off])
                                 : (s[threadIdx.x] + s[threadIdx.x + off]);
    __syncthreads();
  }
  if (threadIdx.x == 0) out[0] = s[0];
}

__global__ void __launch_bounds__(TPB)
logsoftmax_write_kernel(const float* __restrict__ x, int n,
                        const float* __restrict__ maxp, const float* __restrict__ sump,
                        float* __restrict__ o1, float* __restrict__ o2) {
  float mx = maxp[0];
  float ls = logf(sump[0]);
  for (int i = blockIdx.x * TPB + threadIdx.x; i < n; i += gridDim.x * TPB) {
    float v = x[i] - mx - ls;
    o1[i] = v;
    if (o2) o2[i] = v;
  }
}

// ---------------------------------------------------------------------------
// attn_applied[j] = sum_l w[l] * E[l,j]   (split-K partials, deterministic)
// grid = ACH blocks; thread t handles columns 4t..4t+3 (v4f), 32 rows/chunk.
// ---------------------------------------------------------------------------
__global__ void __launch_bounds__(TPB)
attn_apply_partial_kernel(const float* __restrict__ w, const float* __restrict__ E,
                          float* __restrict__ partial) {
  int c = blockIdx.x;
  int t = threadIdx.x;
  v4f acc = {};
  const int rows = Ldim / ACH;              // 32
  int l0 = c * rows;
#pragma unroll 4
  for (int l = l0; l < l0 + rows; ++l) {
    float wl = w[l];
    v4f e = *((const v4f*)(E + (size_t)l * Hdim) + t);
    acc.x += wl * e.x; acc.y += wl * e.y; acc.z += wl * e.z; acc.w += wl * e.w;
  }
  *((v4f*)(partial + (size_t)c * Hdim) + t) = acc;
}

__global__ void __launch_bounds__(TPB)
attn_apply_reduce_kernel(const float* __restrict__ partial, float* __restrict__ cat2) {
  int t = threadIdx.x;                       // one block of 256, 4 cols each
  v4f acc = {};
  for (int c = 0; c < ACH; ++c) {
    v4f p = *((const v4f*)(partial + (size_t)c * Hdim) + t);
    acc.x += p.x; acc.y += p.y; acc.z += p.z; acc.w += p.w;
  }
  *((v4f*)(cat2 + Hdim) + t) = acc;          // cat2[0:H] already holds emb
}

// ---------------------------------------------------------------------------
// LSTM cell pointwise update (torch gate order i,f,g,o), in-place h/c.
// ---------------------------------------------------------------------------
__global__ void __launch_bounds__(TPB)
lstm_update_kernel(const float* __restrict__ gates, float* __restrict__ h,
                   float* __restrict__ c, float* __restrict__ hout,
                   float* __restrict__ cout) {
  int i = blockIdx.x * TPB + threadIdx.x;
  if (i >= Hdim) return;
  float ig = gates[i];
  float fg = gates[Hdim + i];
  float gg = gates[2 * Hdim + i];
  float og = gates[3 * Hdim + i];
  float si = 1.0f / (1.0f + expf(-ig));
  float sf = 1.0f / (1.0f + expf(-fg));
  float so = 1.0f / (1.0f + expf(-og));
  float c2 = sf * c[i] + si * tanhf(gg);
  float h2 = so * tanhf(c2);
  h[i] = h2;
  c[i] = c2;
  if (hout) { hout[i] = h2; cout[i] = c2; }
}

// ---------------------------------------------------------------------------
extern "C" void kernel_launch(void* const* d_in, const int* in_sizes, int n_in,
                              void* d_out, int out_size, void* d_ws, size_t ws_size,
                              hipStream_t stream) {
  const int*   tok    = (const int*)d_in[0];
  const float* h0     = (const float*)d_in[1];
  const float* c0     = (const float*)d_in[2];
  const float* enc    = (const float*)d_in[3];
  const float* embT   = (const float*)d_in[4];
  const float* attn_W = (const float*)d_in[5];
  const float* attn_b = (const float*)d_in[6];
  const float* comb_W = (const float*)d_in[7];
  const float* comb_b = (const float*)d_in[8];
  const float* out_W  = (const float*)d_in[9];
  const float* out_b  = (const float*)d_in[10];
  const float* W_ih[3] = {(const float*)d_in[11], (const float*)d_in[15], (const float*)d_in[19]};
  const float* W_hh[3] = {(const float*)d_in[12], (const float*)d_in[16], (const float*)d_in[20]};
  const float* b_ih[3] = {(const float*)d_in[13], (const float*)d_in[17], (const float*)d_in[21]};
  const float* b_hh[3] = {(const float*)d_in[14], (const float*)d_in[18], (const float*)d_in[22]};

  float* ws = (float*)d_ws;
  float* cat1    = ws + 0;        // 2048
  float* scores  = ws + 2048;     // 2048
  float* attnw   = ws + 4096;     // 2048
  float* cat2    = ws + 6144;     // 2048
  float* xbuf    = ws + 8192;     // 1024
  float* hbuf    = ws + 9216;     // 1024
  float* cbuf    = ws + 10240;    // 1024
  float* gates   = ws + 11264;    // 4096
  float* scrA    = ws + 15360;    // union: gemv partials (<=2*50257) / attn partials (64*1024)
  float* logits  = ws + 115904;   // 50257
  float* blkred  = ws + 166176;   // 64
  float* gmax    = ws + 166240;   // 1
  float* gsum    = ws + 166241;   // 1
  (void)ws_size; (void)in_sizes; (void)n_in; (void)out_size;

  float* out     = (float*)d_out;
  float* logp_o  = out;                    // V
  float* h_o     = out + Vdim;             // H
  float* c_o     = out + Vdim + Hdim;      // H
  float* aw_o    = out + Vdim + 2 * Hdim;  // L

  // split-K GEMV: partial pass + finalize
  auto run_gemv = [&](const float* W1, const float* x1,
                      const float* W2, const float* x2,
                      const float* b1, const float* b2,
                      float* y, int M, int K, int S, int relu) {
    int nRB = (M + 127) / 128;              // 8 waves x 16 rows per block
    dim3 grid(nRB, S);
    gemv_wmma_partial<<<grid, TPB, 0, stream>>>(W1, x1, W2, x2, scrA, M, K, K / S);
    gemv_finalize_kernel<<<(M + TPB - 1) / TPB, TPB, 0, stream>>>(scrA, b1, b2, y, M, S, relu);
  };

  // 1. embedding gather + concat
  prep_kernel<<<(Hdim + TPB - 1) / TPB, TPB, 0, stream>>>(
      tok, embT, h0, c0, cat1, cat2, hbuf, cbuf);

  // 2. attention scores: attn_W[2048,2048] @ cat1       (S=4 -> 512 waves)
  run_gemv(attn_W, cat1, nullptr, nullptr, attn_b, nullptr, scores, Ldim, 2048, 4, 0);

  // 3. log_softmax over L -> attnw (ws) and d_out attn slot
  reduce_max_kernel<<<8, TPB, 0, stream>>>(scores, Ldim, blkred);
  reduce_final_kernel<<<1, TPB, 0, stream>>>(blkred, 8, gmax, 0);
  reduce_sumexp_kernel<<<8, TPB, 0, stream>>>(scores, Ldim, gmax, blkred);
  reduce_final_kernel<<<1, TPB, 0, stream>>>(blkred, 8, gsum, 1);
  logsoftmax_write_kernel<<<8, TPB, 0, stream>>>(scores, Ldim, gmax, gsum, attnw, aw_o);

  // 4. attn_applied = attnw @ encoder_outputs  -> cat2[H:2H]
  attn_apply_partial_kernel<<<ACH, TPB, 0, stream>>>(attnw, enc, scrA);
  attn_apply_reduce_kernel<<<1, TPB, 0, stream>>>(scrA, cat2);

  // 5. x = relu(comb_W @ cat2 + comb_b)                 (S=8 -> 512 waves)
  run_gemv(comb_W, cat2, nullptr, nullptr, comb_b, nullptr, xbuf, Hdim, 2048, 8, 1);

  // 6. three LSTM cells (cell 1 input = xbuf; cells 2,3 input = h)
  for (int li = 0; li < 3; ++li) {
    const float* xin = (li == 0) ? xbuf : hbuf;
    run_gemv(W_ih[li], xin, W_hh[li], hbuf, b_ih[li], b_hh[li],
             gates, 4 * Hdim, Hdim, 8, 0);              // S=8 -> 2048 waves
    bool last = (li == 2);
    lstm_update_kernel<<<(Hdim + TPB - 1) / TPB, TPB, 0, stream>>>(
        gates, hbuf, cbuf, last ? h_o : nullptr, last ? c_o : nullptr);
  }

  // 7. logits = out_W @ h + out_b  (205.9 MB stream)    (S=2 -> 6284 waves)
  run_gemv(out_W, hbuf, nullptr, nullptr, out_b, nullptr, logits, Vdim, Hdim, 2, 0);

  // 8. log_softmax over V -> d_out
  reduce_max_kernel<<<64, TPB, 0, stream>>>(logits, Vdim, blkred);
  reduce_final_kernel<<<1, TPB, 0, stream>>>(blkred, 64, gmax, 0);
  reduce_sumexp_kernel<<<64, TPB, 0, stream>>>(logits, Vdim, gmax, blkred);
  reduce_final_kernel<<<1, TPB, 0, stream>>>(blkred, 64, gsum, 1);
  logsoftmax_write_kernel<<<64, TPB, 0, stream>>>(logits, Vdim, gmax, gsum, logp_o, nullptr);
}